// GraphAttentionConv_69930657513679
// MI455X (gfx1250) — compile-verified
//
#include <hip/hip_runtime.h>
#include <hip/hip_bf16.h>
#include <math.h>
#include <stdint.h>

typedef float v2f __attribute__((ext_vector_type(2)));
typedef float v8f __attribute__((ext_vector_type(8)));

#define BATCH 4
#define NNODE 256
#define CDIM  256
#define NH    8
#define NF    64
#define HF    512          // NH*NF
#define BN    1024         // BATCH*NNODE
#define ITILE 4            // i-rows per block in kernel B

// ---------------------------------------------------------------------------
// Kernel A: g_l = x @ W_l ; g_r = x @ W_r    (M=1024, N=512, K=256, fp32 WMMA)
// One wave computes one 16x16 output tile via V_WMMA_F32_16X16X4_F32.
// grid = 512 blocks * 8 waves = 4096 waves = 2 matrices * 64 Mtiles * 32 Ntiles
// ---------------------------------------------------------------------------
__global__ __launch_bounds__(256) void gat_proj_gemm(
    const float* __restrict__ x,
    const float* __restrict__ Wl,
    const float* __restrict__ Wr,
    float* __restrict__ gl,
    float* __restrict__ gr) {
  const int wave = threadIdx.x >> 5;
  const int lane = threadIdx.x & 31;
  const int lo = lane & 15;        // M (A) / N (B,C) index within tile
  const int hi = lane >> 4;        // K sub-phase selector
  const unsigned gw = blockIdx.x * 8u + wave;
  const int mat = (int)(gw >> 11);           // 0 -> W_l, 1 -> W_r
  const unsigned t = gw & 2047u;
  const int m0 = (int)(t >> 5) * 16;         // 64 M tiles
  const int n0 = (int)(t & 31u) * 16;        // 32 N tiles
  const float* __restrict__ W = mat ? Wr : Wl;
  float* __restrict__ g = mat ? gr : gl;

  v8f acc = {0.f, 0.f, 0.f, 0.f, 0.f, 0.f, 0.f, 0.f};
  const float* arow = x + (m0 + lo) * CDIM;
  const float* bcol = W + n0 + lo;
  for (int k0 = 0; k0 < CDIM; k0 += 4) {
    const int ka = k0 + 2 * hi;              // even -> 8B aligned
    v2f afrag = *(const v2f*)(arow + ka);    // A[m][ka], A[m][ka+1]
    v2f bfrag;
    bfrag.x = bcol[(size_t)ka * HF];         // B[ka][n]
    bfrag.y = bcol[(size_t)(ka + 1) * HF];   // B[ka+1][n]
    acc = __builtin_amdgcn_wmma_f32_16x16x4_f32(
        false, afrag, false, bfrag, (short)0, acc, false, false);
  }
  float* ocol = g + (size_t)m0 * HF + n0 + lo;
#pragma unroll
  for (int r = 0; r < 8; ++r)
    ocol[(size_t)(r + 8 * hi) * HF] = acc[r];
}

// ---------------------------------------------------------------------------
// Kernel B: e[b,i,j,h] = sum_f lrelu(g_r[b,i,h,f] + g_l[b,j,h,f]) * attn_w[f]
//           mask with adj[i,j,h]; softmax over h (8 lanes, in-wave shfl_xor);
//           write a in [b][h][i][j] layout for the aggregation GEMM.
// Each block: one b, a tile of 4 i-rows (amortizes every g_l load over 4
// accumulators -> 4x less L2 traffic), and half of the j range (keeps 4096
// waves in flight). g_r rows staged into LDS with CDNA5 async b128 copies;
// [4][8][68] padding => heads on distinct LDS banks, float4-aligned rows.
// ---------------------------------------------------------------------------
__global__ __launch_bounds__(256) void gat_attn_softmax(
    const float* __restrict__ gl,
    const float* __restrict__ gr,
    const float* __restrict__ aw,
    const int* __restrict__ adj,
    float* __restrict__ a_out) {
  __shared__ float s_gri[ITILE][NH][NF + 4];
  __shared__ float s_aw[NF];
  const int bid = blockIdx.x;
  const int jhalf = bid & 1;                  // which half of j-range
  const int i0 = ((bid >> 1) & 63) * ITILE;   // first of 4 i-rows
  const int b = bid >> 7;
  const int tid = threadIdx.x;

  // ---- stage g_r[b, i0..i0+3, :, :] into LDS via async global->LDS b128 ----
  {
    const float* base = gr + ((size_t)b * NNODE + i0) * HF;
#pragma unroll
    for (int c = tid; c < (ITILE * HF) / 4; c += 256) {   // 512 b128 chunks
      const int ii = c >> 7;
      const int rem = c & 127;
      const int h2 = rem >> 4;
      const int fq = rem & 15;
      const float* gsrc = base + ii * HF + h2 * NF + fq * 4;
      const unsigned ldsoff = (unsigned)(uintptr_t)(&s_gri[ii][h2][fq * 4]);
      asm volatile("global_load_async_to_lds_b128 %0, %1, off"
                   :: "v"(ldsoff), "v"(gsrc)
                   : "memory");
    }
    if (tid < NF) s_aw[tid] = aw[tid];
    asm volatile("s_wait_asynccnt 0" ::: "memory");
  }
  __syncthreads();

  const int h = tid & 7;
  const int jl = tid >> 3;                       // 0..31
  const int jbase = jhalf * (NNODE / 2);

  for (int jt = 0; jt < NNODE / 2; jt += 32) {
    const int j = jbase + jt + jl;
    const float* glrow = gl + ((size_t)b * NNODE + j) * HF + h * NF;
    float s0 = 0.f, s1 = 0.f, s2 = 0.f, s3 = 0.f;
#pragma unroll
    for (int f = 0; f < NF; f += 4) {
      const float4 gv = *(const float4*)(glrow + f);   // shared by 4 i-rows
      const float w0 = s_aw[f + 0], w1 = s_aw[f + 1];
      const float w2 = s_aw[f + 2], w3 = s_aw[f + 3];
#pragma unroll
      for (int ii = 0; ii < ITILE; ++ii) {
        float t0 = s_gri[ii][h][f + 0] + gv.x;
        float t1 = s_gri[ii][h][f + 1] + gv.y;
        float t2 = s_gri[ii][h][f + 2] + gv.z;
        float t3 = s_gri[ii][h][f + 3] + gv.w;
        t0 = fmaxf(t0, 0.2f * t0);                     // leaky_relu(0.2)
        t1 = fmaxf(t1, 0.2f * t1);
        t2 = fmaxf(t2, 0.2f * t2);
        t3 = fmaxf(t3, 0.2f * t3);
        float acc = (ii == 0) ? s0 : (ii == 1) ? s1 : (ii == 2) ? s2 : s3;
        acc = fmaf(t0, w0, acc);
        acc = fmaf(t1, w1, acc);
        acc = fmaf(t2, w2, acc);
        acc = fmaf(t3, w3, acc);
        if (ii == 0) s0 = acc; else if (ii == 1) s1 = acc;
        else if (ii == 2) s2 = acc; else s3 = acc;
      }
    }
#pragma unroll
    for (int ii = 0; ii < ITILE; ++ii) {
      const float s = (ii == 0) ? s0 : (ii == 1) ? s1 : (ii == 2) ? s2 : s3;
      const int i = i0 + ii;
      const int ad = adj[((size_t)i * NNODE + j) * NH + h];
      const float e = ad ? s : -__builtin_inff();
      // softmax over the 8 heads: lanes jl*8+h, reduce within groups of 8
      float m = e;
      for (int o = 4; o >= 1; o >>= 1)
        m = fmaxf(m, __shfl_xor(m, o, 8));
      const float p = ad ? __expf(e - m) : 0.f;  // >=1 head unmasked per (i,j)
      float sum = p;
      for (int o = 4; o >= 1; o >>= 1)
        sum += __shfl_xor(sum, o, 8);
      a_out[(((size_t)b * NH + h) * NNODE + i) * NNODE + j] = p / sum;
    }
  }
}

// ---------------------------------------------------------------------------
// Kernel C: out[b,i,h,f] = sum_j a[b,h,i,j] * g_r[b,j,h,f]
// 32 batched GEMMs (b,h): M=256, N=64, K=256, fp32 WMMA 16x16x4.
// grid = 256 blocks * 8 waves = 2048 waves = 32 batches * 16 Mtiles * 4 Ntiles
// ---------------------------------------------------------------------------
__global__ __launch_bounds__(256) void gat_aggregate_gemm(
    const float* __restrict__ a_ws,
    const float* __restrict__ gr,
    float* __restrict__ out) {
  const int wave = threadIdx.x >> 5;
  const int lane = threadIdx.x & 31;
  const int lo = lane & 15;
  const int hi = lane >> 4;
  const unsigned gw = blockIdx.x * 8u + wave;
  const int bh = (int)(gw >> 6);               // 0..31
  const int b = bh >> 3, h = bh & 7;
  const unsigned t = gw & 63u;
  const int m0 = (int)(t >> 2) * 16;           // 16 M tiles
  const int n0 = (int)(t & 3u) * 16;           // 4 N tiles

  const float* __restrict__ A  = a_ws + (size_t)bh * NNODE * NNODE;     // [256][256]
  const float* __restrict__ Bm = gr  + (size_t)b * NNODE * HF + h * NF; // [256][64], ld=512
  float* __restrict__ O        = out + (size_t)b * NNODE * HF + h * NF; // ld=512

  v8f acc = {0.f, 0.f, 0.f, 0.f, 0.f, 0.f, 0.f, 0.f};
  const float* arow = A + (m0 + lo) * NNODE;
  const float* bcol = Bm + n0 + lo;
  for (int k0 = 0; k0 < NNODE; k0 += 4) {
    const int ka = k0 + 2 * hi;
    v2f afrag = *(const v2f*)(arow + ka);
    v2f bfrag;
    bfrag.x = bcol[(size_t)ka * HF];
    bfrag.y = bcol[(size_t)(ka + 1) * HF];
    acc = __builtin_amdgcn_wmma_f32_16x16x4_f32(
        false, afrag, false, bfrag, (short)0, acc, false, false);
  }
  float* ocol = O + (size_t)m0 * HF + n0 + lo;
#pragma unroll
  for (int r = 0; r < 8; ++r)
    ocol[(size_t)(r + 8 * hi) * HF] = acc[r];
}

// ---------------------------------------------------------------------------
extern "C" void kernel_launch(void* const* d_in, const int* in_sizes, int n_in,
                              void* d_out, int out_size, void* d_ws, size_t ws_size,
                              hipStream_t stream) {
  const float* x   = (const float*)d_in[0];   // [4,256,256]
  const float* Wl  = (const float*)d_in[1];   // [256,512]
  const float* Wr  = (const float*)d_in[2];   // [256,512]
  const float* aw  = (const float*)d_in[3];   // [64]
  const int*   adj = (const int*)d_in[4];     // [256,256,8]
  float* out = (float*)d_out;                 // [4,256,512]

  float* gl   = (float*)d_ws;                 // 2 MB
  float* gr   = gl + (size_t)BN * HF;         // 2 MB
  float* a_ws = gr + (size_t)BN * HF;         // 8 MB, layout [b][h][i][j]

  gat_proj_gemm<<<512, 256, 0, stream>>>(x, Wl, Wr, gl, gr);
  gat_attn_softmax<<<BATCH * (NNODE / ITILE) * 2, 256, 0, stream>>>(gl, gr, aw, adj, a_ws);
  gat_aggregate_gemm<<<256, 256, 0, stream>>>(a_ws, gr, out);
}